// FM_Layer_919123001694
// MI455X (gfx1250) — compile-verified
//
#include <hip/hip_runtime.h>

#define N_DENSE  13
#define N_SPARSE 26
#define VOCAB    100000
#define LATENT   16
#define FEAT_LEN (N_DENSE + (long)N_SPARSE * VOCAB)   // 2,600,013

typedef __attribute__((ext_vector_type(2))) float v2f;
typedef __attribute__((ext_vector_type(8))) float v8f;

// One wave32 handles 2 samples: lanes [0..15] = sample b0 (lane = latent dim),
// lanes [16..31] = sample b0+1. Dense part computed with fp32 WMMA 16x16x4
// (K padded 13->16, 4 k-slices, two accumulators: xV and x2V2).
__global__ __launch_bounds__(64) void fm_fused_kernel(
    const float* __restrict__ dense, const int* __restrict__ sparse,
    const float* __restrict__ w0p,   const float* __restrict__ w,
    const float* __restrict__ V,     float* __restrict__ out, int nBatch)
{
  const int lane   = threadIdx.x & 31;
  const int l      = lane & 15;              // latent index within half-wave
  const int half   = lane >> 4;              // which sample of the pair
  const int waveId = (blockIdx.x * blockDim.x + threadIdx.x) >> 5;
  const int b0     = waveId * 2;
  const int b      = b0 + half;
  const bool valid = (b < nBatch);

  // ---------- sparse indices: 13 x b64 vector loads (row is 8B-aligned) ----------
  const int srow = valid ? b * N_SPARSE : 0;
  int sidx[N_SPARSE];
  {
    const int2* sp = (const int2*)(sparse + srow);
#pragma unroll
    for (int p = 0; p < N_SPARSE / 2; ++p) {
      int2 v = sp[p];
      sidx[2 * p]     = v.x;
      sidx[2 * p + 1] = v.y;
    }
  }

  // ---------- sparse second-order gathers: lane l walks latent row l ----------
  const float* Vrow = V + (long)l * FEAT_LEN;
  float acc = 0.f, acc2 = 0.f, firstAcc = 0.f;
#pragma unroll
  for (int s = 0; s < N_SPARSE; ++s) {
    int   g = N_DENSE + s * VOCAB + sidx[s];
    float v = Vrow[g];
    acc  += v;
    acc2 += v * v;
  }
  // first-order: lane l covers sparse fields l and l+16, plus dense term d=l
  firstAcc += w[N_DENSE + l * VOCAB + sidx[l]];
  if (l + 16 < N_SPARSE) {
    int s = l + 16;
    firstAcc += w[N_DENSE + s * VOCAB + sidx[s]];
  }
  if (l < N_DENSE)
    firstAcc += dense[(valid ? b : 0) * N_DENSE + l] * w[l];

  // ---------- dense part via V_WMMA_F32_16X16X4_F32 (branchless fragments) ----
  // A: 16x16 tile of dense rows (only rows 0,1 live = this wave's samples),
  // B: 16x16 tile of V^T head (B[k][n] = V[n][k]), K padded 13->16 with zeros.
  const int  m       = l;                      // A row / B column for this lane
  const bool aLive   = (m < 2) && ((b0 + m) < nBatch);
  const int  aRow    = aLive ? (b0 + m) : 0;   // clamped, always in range
  const float amask  = aLive ? 1.f : 0.f;
  float aF[8], bF[8];
#pragma unroll
  for (int kk = 0; kk < 4; ++kk) {
    const int k0 = kk * 4 + half * 2;          // per-lane K pair (ISA 16x4 A layout)
    const int k1 = k0 + 1;
    const int c0 = (k0 < N_DENSE) ? k0 : 0;    // clamped addresses, no branches
    const int c1 = (k1 < N_DENSE) ? k1 : 0;
    const float m0 = (k0 < N_DENSE) ? 1.f : 0.f;
    const float m1 = (k1 < N_DENSE) ? 1.f : 0.f;
    aF[2 * kk]     = amask * m0 * dense[aRow * N_DENSE + c0];
    aF[2 * kk + 1] = amask * m1 * dense[aRow * N_DENSE + c1];
    bF[2 * kk]     = m0 * V[(long)m * FEAT_LEN + c0];   // B[k][m] = V[m][k]
    bF[2 * kk + 1] = m1 * V[(long)m * FEAT_LEN + c1];
  }

  v8f cXV = {0.f,0.f,0.f,0.f,0.f,0.f,0.f,0.f};
  v8f cX2 = {0.f,0.f,0.f,0.f,0.f,0.f,0.f,0.f};
#pragma unroll
  for (int kk = 0; kk < 4; ++kk) {
    v2f A  = {aF[2 * kk], aF[2 * kk + 1]};
    v2f B  = {bF[2 * kk], bF[2 * kk + 1]};
    v2f A2 = {A[0] * A[0], A[1] * A[1]};
    v2f B2 = {B[0] * B[0], B[1] * B[1]};
    cXV = __builtin_amdgcn_wmma_f32_16x16x4_f32(false, A,  false, B,
                                                (short)0, cXV, false, false);
    cX2 = __builtin_amdgcn_wmma_f32_16x16x4_f32(false, A2, false, B2,
                                                (short)0, cX2, false, false);
  }

  // C layout: VGPR0 lanes0-15 = M=0 (sample b0), VGPR1 lanes0-15 = M=1
  // (sample b0+1). Route M=1 values to lanes 16-31.
  float xv1 = __shfl(cXV[1], l, 32);
  float x21 = __shfl(cX2[1], l, 32);
  float xvD = half ? xv1 : cXV[0];
  float x2D = half ? x21 : cX2[0];

  // ---------- FM combine + 16-lane reduction ----------
  float t  = acc  + xvD;
  float t2 = acc2 + x2D;
  float partial = 0.5f * (t * t - t2) + firstAcc;
#pragma unroll
  for (int off = 8; off > 0; off >>= 1)
    partial += __shfl_xor(partial, off, 16);

  if (l == 0 && valid)
    out[b] = partial + w0p[0];
}

extern "C" void kernel_launch(void* const* d_in, const int* in_sizes, int n_in,
                              void* d_out, int out_size, void* d_ws, size_t ws_size,
                              hipStream_t stream) {
  const float* dense  = (const float*)d_in[0];
  const int*   sparse = (const int*)  d_in[1];
  const float* w0     = (const float*)d_in[2];
  const float* w      = (const float*)d_in[3];
  const float* V      = (const float*)d_in[4];
  float*       out    = (float*)d_out;

  const int nBatch = out_size;                 // output is [BATCH, 1]
  const int samplesPerBlock = 4;               // 64 threads = 2 waves * 2 samples
  const int blocks = (nBatch + samplesPerBlock - 1) / samplesPerBlock;
  fm_fused_kernel<<<blocks, 64, 0, stream>>>(dense, sparse, w0, w, V, out, nBatch);

  (void)in_sizes; (void)n_in; (void)d_ws; (void)ws_size;
}